// GCN_58411555225950
// MI455X (gfx1250) — compile-verified
//
#include <hip/hip_runtime.h>
#include <hip/hip_bf16.h>

// ---------------------------------------------------------------------------
// GCN forward on gfx1250 (MI455X).
//   Projections use V_WMMA_F32_16X16X4_F32 (exact fp32 WMMA, wave32).
//   Edge aggregation uses coalesced float4 gathers + global f32 atomics
//   (both intermediate matrices fit the 192MB L2, so atomics stay on-chip).
// ---------------------------------------------------------------------------

typedef __attribute__((ext_vector_type(2))) float v2f;
typedef __attribute__((ext_vector_type(8))) float v8f;

#define N_NODES   100000
#define N_EDGES   1600000
#define IN_FEATS  128
#define H_FEATS   64
#define N_CLASSES 16

// ------------------------------ utility kernels ----------------------------

__global__ void zero_f32(float* __restrict__ p, long long n) {
    long long i = (long long)blockIdx.x * blockDim.x + threadIdx.x;
    long long stride = (long long)gridDim.x * blockDim.x;
    for (; i < n; i += stride) p[i] = 0.0f;
}

__global__ void init_out(float* __restrict__ out, const float* __restrict__ b2) {
    int i = threadIdx.x;
    if (i < N_CLASSES) out[i] = b2[i];
}

__global__ void degree_kernel(const int* __restrict__ src, const int* __restrict__ dst,
                              float* __restrict__ deg_out, float* __restrict__ deg_in) {
    int e = blockIdx.x * blockDim.x + threadIdx.x;
    if (e < N_EDGES) {
        atomicAdd(&deg_out[src[e]], 1.0f);
        atomicAdd(&deg_in[dst[e]], 1.0f);
    }
}

// deg -> rsqrt(max(deg,1)) in place, over both arrays (2N contiguous floats)
__global__ void norm_kernel(float* __restrict__ deg) {
    int i = blockIdx.x * blockDim.x + threadIdx.x;
    if (i < 2 * N_NODES) {
        deg[i] = rsqrtf(fmaxf(deg[i], 1.0f));
    }
}

// ------------------------------ WMMA GEMMs ---------------------------------
// Y[N x 64] = (X[N x 128] * norm_out[:,None]) @ W[128 x 64]
// One wave computes one 16-row tile across all 4 column tiles.
__global__ __launch_bounds__(256) void gemm1_wmma(
    const float* __restrict__ X, const float* __restrict__ norm,
    const float* __restrict__ W, float* __restrict__ Y)
{
    const int wave = blockIdx.x * (blockDim.x >> 5) + (threadIdx.x >> 5);
    const int lane = threadIdx.x & 31;
    const int row0 = wave * 16;
    if (row0 >= N_NODES) return;

    const int m     = lane & 15;        // row within tile (A) / col within tile (B,C,D)
    const int hi    = lane >> 4;        // lane half
    const int khalf = hi * 2;           // K sub-offset for A/B fragments
    const float nrm = norm[row0 + m];
    const float* arow = X + (long long)(row0 + m) * IN_FEATS;

    v8f acc0 = {}, acc1 = {}, acc2 = {}, acc3 = {};

    for (int k0 = 0; k0 < IN_FEATS; k0 += 4) {
        v2f a;
        a.x = arow[k0 + khalf]     * nrm;
        a.y = arow[k0 + khalf + 1] * nrm;
        const float* w0 = W + (long long)(k0 + khalf) * H_FEATS + m;
        v2f b;
        b.x = w0[0];        b.y = w0[H_FEATS];
        acc0 = __builtin_amdgcn_wmma_f32_16x16x4_f32(false, a, false, b, (short)0, acc0, false, false);
        b.x = w0[16];       b.y = w0[H_FEATS + 16];
        acc1 = __builtin_amdgcn_wmma_f32_16x16x4_f32(false, a, false, b, (short)0, acc1, false, false);
        b.x = w0[32];       b.y = w0[H_FEATS + 32];
        acc2 = __builtin_amdgcn_wmma_f32_16x16x4_f32(false, a, false, b, (short)0, acc2, false, false);
        b.x = w0[48];       b.y = w0[H_FEATS + 48];
        acc3 = __builtin_amdgcn_wmma_f32_16x16x4_f32(false, a, false, b, (short)0, acc3, false, false);
    }

    #pragma unroll
    for (int r = 0; r < 8; ++r) {
        long long row = row0 + r + 8 * hi;
        float* yr = Y + row * H_FEATS + m;
        yr[0]  = acc0[r];
        yr[16] = acc1[r];
        yr[32] = acc2[r];
        yr[48] = acc3[r];
    }
}

// Y[N x 16] = (H[N x 64] * norm_out[:,None]) @ W[64 x 16]
__global__ __launch_bounds__(256) void gemm2_wmma(
    const float* __restrict__ H, const float* __restrict__ norm,
    const float* __restrict__ W, float* __restrict__ Y)
{
    const int wave = blockIdx.x * (blockDim.x >> 5) + (threadIdx.x >> 5);
    const int lane = threadIdx.x & 31;
    const int row0 = wave * 16;
    if (row0 >= N_NODES) return;

    const int m     = lane & 15;
    const int hi    = lane >> 4;
    const int khalf = hi * 2;
    const float nrm = norm[row0 + m];
    const float* arow = H + (long long)(row0 + m) * H_FEATS;

    v8f acc = {};
    for (int k0 = 0; k0 < H_FEATS; k0 += 4) {
        v2f a;
        a.x = arow[k0 + khalf]     * nrm;
        a.y = arow[k0 + khalf + 1] * nrm;
        v2f b;
        b.x = W[(long long)(k0 + khalf)     * N_CLASSES + m];
        b.y = W[(long long)(k0 + khalf + 1) * N_CLASSES + m];
        acc = __builtin_amdgcn_wmma_f32_16x16x4_f32(false, a, false, b, (short)0, acc, false, false);
    }

    #pragma unroll
    for (int r = 0; r < 8; ++r) {
        long long row = row0 + r + 8 * hi;
        Y[row * N_CLASSES + m] = acc[r];
    }
}

// ------------------------------ edge scatter -------------------------------
// AGG[dst] += X[src], 64 feats: E*16 threads, float4 per thread.
__global__ void scatter64(const float* __restrict__ X, const int* __restrict__ src,
                          const int* __restrict__ dst, float* __restrict__ AGG) {
    long long t = (long long)blockIdx.x * blockDim.x + threadIdx.x;
    if (t >= (long long)N_EDGES * 16) return;
    int e = (int)(t >> 4);
    int c = ((int)t & 15) * 4;
    int s = src[e], d = dst[e];
    const float4 v = *(const float4*)(X + (long long)s * H_FEATS + c);
    float* p = AGG + (long long)d * H_FEATS + c;
    atomicAdd(p + 0, v.x);
    atomicAdd(p + 1, v.y);
    atomicAdd(p + 2, v.z);
    atomicAdd(p + 3, v.w);
}

// AGG[dst] += X[src], 16 feats: E*4 threads, float4 per thread.
__global__ void scatter16(const float* __restrict__ X, const int* __restrict__ src,
                          const int* __restrict__ dst, float* __restrict__ AGG) {
    long long t = (long long)blockIdx.x * blockDim.x + threadIdx.x;
    if (t >= (long long)N_EDGES * 4) return;
    int e = (int)(t >> 2);
    int c = ((int)t & 3) * 4;
    int s = src[e], d = dst[e];
    const float4 v = *(const float4*)(X + (long long)s * N_CLASSES + c);
    float* p = AGG + (long long)d * N_CLASSES + c;
    atomicAdd(p + 0, v.x);
    atomicAdd(p + 1, v.y);
    atomicAdd(p + 2, v.z);
    atomicAdd(p + 3, v.w);
}

// H = relu(AGG * norm_in + b1), in place (N x 64)
__global__ void h1_post(float* __restrict__ AGG, const float* __restrict__ norm_in,
                        const float* __restrict__ b1) {
    long long idx = (long long)blockIdx.x * blockDim.x + threadIdx.x;
    if (idx >= (long long)N_NODES * H_FEATS) return;
    int i = (int)(idx >> 6);
    int f = (int)idx & 63;
    float v = AGG[idx] * norm_in[i] + b1[f];
    AGG[idx] = fmaxf(v, 0.0f);
}

// out[j] += mean_i(AGG2[i,j] * norm_in[i])   (b2 already seeded into out)
#define RED_ROWS 512
__global__ __launch_bounds__(256) void reduce_mean(const float* __restrict__ AGG2,
                                                   const float* __restrict__ norm_in,
                                                   float* __restrict__ out) {
    __shared__ float sm[256];
    const int j = threadIdx.x & 15;   // class
    const int r = threadIdx.x >> 4;   // row lane 0..15
    const int base = blockIdx.x * RED_ROWS;
    float s = 0.0f;
    int end = base + RED_ROWS; if (end > N_NODES) end = N_NODES;
    for (int i = base + r; i < end; i += 16)
        s += AGG2[(long long)i * N_CLASSES + j] * norm_in[i];
    sm[threadIdx.x] = s;
    __syncthreads();
    for (int stride = 128; stride >= 16; stride >>= 1) {
        if (threadIdx.x < stride) sm[threadIdx.x] += sm[threadIdx.x + stride];
        __syncthreads();
    }
    if (threadIdx.x < 16)
        atomicAdd(&out[j], sm[threadIdx.x] * (1.0f / (float)N_NODES));
}

// ------------------------------ launcher -----------------------------------

extern "C" void kernel_launch(void* const* d_in, const int* in_sizes, int n_in,
                              void* d_out, int out_size, void* d_ws, size_t ws_size,
                              hipStream_t stream) {
    const float* in_feat = (const float*)d_in[0];
    const int*   src     = (const int*)  d_in[1];
    const int*   dst     = (const int*)  d_in[2];
    const float* W1      = (const float*)d_in[3];
    const float* b1      = (const float*)d_in[4];
    const float* W2      = (const float*)d_in[5];
    const float* b2      = (const float*)d_in[6];
    float* out = (float*)d_out;

    // workspace layout (floats)
    float* ws       = (float*)d_ws;
    float* deg      = ws;                       // [2N] -> becomes norm_out|norm_in
    float* norm_out = ws;                       // [N]
    float* norm_in  = ws + N_NODES;             // [N]
    float* bufA     = ws + 2LL * N_NODES;       // [N*64]  X1, later X2|AGG2
    float* bufB     = bufA + (long long)N_NODES * H_FEATS; // [N*64] AGG1 -> H1
    float* X2       = bufA;                                    // [N*16]
    float* AGG2     = bufA + (long long)N_NODES * N_CLASSES;   // [N*16]

    const int T = 256;

    // 1) zero degree counters, seed out with b2
    zero_f32<<<1024, T, 0, stream>>>(deg, 2LL * N_NODES);
    init_out<<<1, 32, 0, stream>>>(out, b2);

    // 2) degrees
    degree_kernel<<<(N_EDGES + T - 1) / T, T, 0, stream>>>(src, dst, deg, deg + N_NODES);

    // 3) norms in place
    norm_kernel<<<(2 * N_NODES + T - 1) / T, T, 0, stream>>>(deg);

    // 4) zero AGG1
    zero_f32<<<4096, T, 0, stream>>>(bufB, (long long)N_NODES * H_FEATS);

    // 5) X1 = (in_feat * norm_out) @ W1   (WMMA f32 16x16x4)
    {
        int tiles = N_NODES / 16;               // 6250
        int wavesPerBlock = T / 32;             // 8
        int grid = (tiles + wavesPerBlock - 1) / wavesPerBlock;
        gemm1_wmma<<<grid, T, 0, stream>>>(in_feat, norm_out, W1, bufA);
    }

    // 6) AGG1[dst] += X1[src]
    {
        long long work = (long long)N_EDGES * 16;
        scatter64<<<(int)((work + T - 1) / T), T, 0, stream>>>(bufA, src, dst, bufB);
    }

    // 7) H1 = relu(AGG1 * norm_in + b1)
    {
        long long work = (long long)N_NODES * H_FEATS;
        h1_post<<<(int)((work + T - 1) / T), T, 0, stream>>>(bufB, norm_in, b1);
    }

    // 8) X2 = (H1 * norm_out) @ W2   (WMMA f32 16x16x4)
    {
        int tiles = N_NODES / 16;
        int wavesPerBlock = T / 32;
        int grid = (tiles + wavesPerBlock - 1) / wavesPerBlock;
        gemm2_wmma<<<grid, T, 0, stream>>>(bufB, norm_out, W2, X2);
    }

    // 9) zero AGG2 (X1 region is dead now)
    zero_f32<<<1024, T, 0, stream>>>(AGG2, (long long)N_NODES * N_CLASSES);

    // 10) AGG2[dst] += X2[src]
    {
        long long work = (long long)N_EDGES * 4;
        scatter16<<<(int)((work + T - 1) / T), T, 0, stream>>>(X2, src, dst, AGG2);
    }

    // 11) out = mean_i(AGG2[i]*norm_in[i]) + b2
    reduce_mean<<<(N_NODES + RED_ROWS - 1) / RED_ROWS, T, 0, stream>>>(AGG2, norm_in, out);
}